// ParallelMagLoss_20718922236693
// MI455X (gfx1250) — compile-verified
//
#include <hip/hip_runtime.h>
#include <cstdint>
#include <cstddef>

typedef __attribute__((ext_vector_type(4))) float v4f;

namespace {
constexpr int B   = 256;        // batch rows
constexpr int C   = 100000;     // classes
constexpr int S   = 10;         // column splits per row
constexpr int CS  = C / S;      // 10000 columns per block (40 KB in LDS)
constexpr int NV4 = CS / 4;     // 2500 float4 per chunk
constexpr int NOH = (CS - 4) / 4; // 2499 aligned float4 one_hot stores per chunk
constexpr int NT  = 256;        // threads per block (8 wave32)
}

// ---------------------------------------------------------------------------
// Kernel A: per (row, split) block — async-stage chunk into LDS, write the
// one_hot chunk with 16B non-temporal stores while the copy is in flight,
// then two-pass (max, sum-exp) softmax partials over the LDS-resident chunk.
// ---------------------------------------------------------------------------
__global__ __launch_bounds__(NT)
void magloss_partial(const float* __restrict__ ct,
                     const float* __restrict__ ctm,
                     const int*   __restrict__ target,
                     float*       __restrict__ one_hot,
                     float*       __restrict__ pm,
                     float*       __restrict__ ps)
{
    __shared__ __align__(16) float tile[CS];
    __shared__ float red[8];

    const int tid   = threadIdx.x;
    const int split = blockIdx.x;
    const int row   = blockIdx.y;
    const int c0    = split * CS;
    const int tgt   = target[row];
    const size_t rowbase = (size_t)row * (size_t)C;

    // ---- async copy: global chunk -> LDS (CDNA5 ASYNCcnt path) ----
    const unsigned lds0 = (unsigned)(size_t)(void*)tile;   // low 32b = LDS offset
    const float* gsrc = ct + rowbase + (size_t)c0;         // 16B aligned chunk
    for (int i = tid; i < NV4; i += NT) {
        unsigned loff = lds0 + (unsigned)i * 16u;
        unsigned long long ga = (unsigned long long)(size_t)(gsrc + 4 * i);
        asm volatile("global_load_async_to_lds_b128 %0, %1, off"
                     :: "v"(loff), "v"(ga) : "memory");
    }

    // ---- overlap with the async copy: stream one_hot (zeros, 1 at target).
    // one_hot = d_out + 2 is only 8B-aligned; chunk base is misaligned by
    // exactly 2 floats, so: 2-float head, 2499 aligned b128 NT stores, 2-float
    // tail. NT keeps the 192MB L2 for the cos_theta read stream.
    {
        float* oh = one_hot + rowbase + (size_t)c0;
        v4f* oh4 = (v4f*)(oh + 2);                          // 16B aligned
        for (int i = tid; i < NOH; i += NT) {
            const int j = c0 + 2 + 4 * i;
            v4f v;
            v.x = (j     == tgt) ? 1.0f : 0.0f;
            v.y = (j + 1 == tgt) ? 1.0f : 0.0f;
            v.z = (j + 2 == tgt) ? 1.0f : 0.0f;
            v.w = (j + 3 == tgt) ? 1.0f : 0.0f;
            __builtin_nontemporal_store(v, oh4 + i);
        }
        if (tid == 0) {
            oh[0]      = (c0          == tgt) ? 1.0f : 0.0f;
            oh[1]      = (c0 + 1      == tgt) ? 1.0f : 0.0f;
            oh[CS - 2] = (c0 + CS - 2 == tgt) ? 1.0f : 0.0f;
            oh[CS - 1] = (c0 + CS - 1 == tgt) ? 1.0f : 0.0f;
        }
    }

    asm volatile("s_wait_asynccnt 0" ::: "memory");
    __syncthreads();

    // blend: substitute the target column with cos_theta_m[row, tgt]
    if (tid == 0 && tgt >= c0 && tgt < c0 + CS)
        tile[tgt - c0] = ctm[rowbase + (size_t)tgt];
    __syncthreads();

    const v4f* t4  = (const v4f*)tile;
    const int wave = tid >> 5, lane = tid & 31;

    // ---- pass 1: block max over the chunk ----
    float m = -__builtin_huge_valf();
    for (int i = tid; i < NV4; i += NT) {
        v4f v = t4[i];
        m = fmaxf(m, fmaxf(fmaxf(v.x, v.y), fmaxf(v.z, v.w)));
    }
    for (int o = 16; o > 0; o >>= 1) m = fmaxf(m, __shfl_xor(m, o, 32));
    if (lane == 0) red[wave] = m;
    __syncthreads();
    m = red[0];
#pragma unroll
    for (int w = 1; w < 8; ++w) m = fmaxf(m, red[w]);
    __syncthreads();

    // ---- pass 2: sum of exp(v - m), exactly 1 exp per element ----
    float ssum = 0.0f;
    for (int i = tid; i < NV4; i += NT) {
        v4f v = t4[i];
        ssum += __expf(v.x - m) + __expf(v.y - m)
              + __expf(v.z - m) + __expf(v.w - m);
    }
    for (int o = 16; o > 0; o >>= 1) ssum += __shfl_xor(ssum, o, 32);
    if (lane == 0) red[wave] = ssum;
    __syncthreads();
    if (tid == 0) {
        float s = 0.0f;
#pragma unroll
        for (int w = 0; w < 8; ++w) s += red[w];
        pm[row * S + split] = m;
        ps[row * S + split] = s;
    }
}

// ---------------------------------------------------------------------------
// Kernel B: merge the S partials per row (LSE merge), compute per-row loss and
// the x_norm regularizer, reduce both means into d_out[0], d_out[1].
// ---------------------------------------------------------------------------
__global__ __launch_bounds__(B)
void magloss_final(const float* __restrict__ pm,
                   const float* __restrict__ ps,
                   const float* __restrict__ ctm,
                   const int*   __restrict__ target,
                   const float* __restrict__ x_norm,
                   float*       __restrict__ out)
{
    __shared__ float rl[8], rg[8];
    const int row = threadIdx.x;

    float m = -__builtin_huge_valf();
#pragma unroll
    for (int s = 0; s < S; ++s) m = fmaxf(m, pm[row * S + s]);
    float sum = 0.0f;
#pragma unroll
    for (int s = 0; s < S; ++s) sum += ps[row * S + s] * __expf(pm[row * S + s] - m);

    const int tgt = target[row];
    float loss = m + __logf(sum) - ctm[(size_t)row * (size_t)C + (size_t)tgt];

    const float xn = x_norm[row];
    float g = xn * (1.0f / (110.0f * 110.0f)) + 1.0f / xn;

    for (int o = 16; o > 0; o >>= 1) {
        loss += __shfl_xor(loss, o, 32);
        g    += __shfl_xor(g,    o, 32);
    }
    const int wave = row >> 5, lane = row & 31;
    if (lane == 0) { rl[wave] = loss; rg[wave] = g; }
    __syncthreads();
    if (row == 0) {
        float L = 0.0f, G = 0.0f;
#pragma unroll
        for (int w = 0; w < 8; ++w) { L += rl[w]; G += rg[w]; }
        out[0] = L * (1.0f / (float)B);
        out[1] = G * (1.0f / (float)B);
    }
}

// ---------------------------------------------------------------------------
extern "C" void kernel_launch(void* const* d_in, const int* in_sizes, int n_in,
                              void* d_out, int out_size, void* d_ws, size_t ws_size,
                              hipStream_t stream) {
    (void)in_sizes; (void)n_in; (void)out_size; (void)ws_size;
    const float* ct  = (const float*)d_in[0];   // cos_theta   [B,C] f32
    const float* ctm = (const float*)d_in[1];   // cos_theta_m [B,C] f32
    const float* xn  = (const float*)d_in[2];   // x_norm      [B]   f32
    const int*   tgt = (const int*)d_in[3];     // target      [B]   i32

    float* out     = (float*)d_out;             // [loss, loss_g, one_hot...]
    float* one_hot = out + 2;

    float* pm = (float*)d_ws;                   // [B*S] partial maxima
    float* ps = pm + B * S;                     // [B*S] partial sums

    dim3 grid(S, B);
    magloss_partial<<<grid, NT, 0, stream>>>(ct, ctm, tgt, one_hot, pm, ps);
    magloss_final<<<1, B, 0, stream>>>(pm, ps, ctm, tgt, xn, out);
}